// GuidedAttentionLoss_231928234108
// MI455X (gfx1250) — compile-verified
//
#include <hip/hip_runtime.h>
#include <hip/hip_bf16.h>

// GuidedAttentionLoss for MI455X (gfx1250, wave32).
// out = mean(A * guided_mask), A: [B=64, T=2000, E=400] f32.
// Key facts exploited:
//  * rows with i >= out_len[b] contribute exactly 0 -> skipped (no loads)
//  * exp(-d*d*3.125) == 0.0f exactly in f32 for |d| >= 6, so a 32-wide band
//    (lane j = ideal_j-15 .. ideal_j+16) captures every nonzero term exactly.
//  * accumulation runs on the matrix pipe: V_WMMA_F32_16X16X4_F32 with an
//    all-ones B matrix sums 64 lane-products per issue into the f32 C tile
//    (sum(C) == 16 * sum(inputs), exact since x*1.0 adds are plain RNE adds).

typedef __attribute__((ext_vector_type(2))) float v2f;
typedef __attribute__((ext_vector_type(8))) float v8f;

#define B_DIM 64
#define T_DIM 2000
#define E_DIM 400
#define INV_2SIG2 3.125f            // 1 / (2 * 0.4^2)
#define ROWS_PER_WAVE 16
#define WAVES_PER_BLK 8
#define ROWS_PER_BLK (ROWS_PER_WAVE * WAVES_PER_BLK)   // 128
#define GRID_X ((T_DIM + ROWS_PER_BLK - 1) / ROWS_PER_BLK) // 16
#define NPART (GRID_X * B_DIM)                         // 1024
#define N_TOTAL 51200000.0f                            // 64*2000*400

__global__ __launch_bounds__(256)
void ga_band_sum_kernel(const float* __restrict__ aw,
                        const int* __restrict__ in_len,
                        const int* __restrict__ out_len,
                        float* __restrict__ partials)
{
    const int lane = threadIdx.x & 31;
    const int wave = threadIdx.x >> 5;
    const int b    = blockIdx.y;
    const int i_base = blockIdx.x * ROWS_PER_BLK + wave * ROWS_PER_WAVE;

    const int   il   = in_len[b];
    const int   ol   = out_len[b];
    const float fin  = (float)il;
    const float fout = (float)ol;
    const int   imax = (ol < T_DIM) ? ol : T_DIM;   // rows >= out_len are all-zero

    const float* __restrict__ row0 = aw + (size_t)b * T_DIM * E_DIM;

    v8f c = {};                    // WMMA f32 accumulator tile (16x16 / 8 VGPRs)
    v2f ones; ones.x = 1.0f; ones.y = 1.0f;   // B matrix: all ones (4x16)

    #pragma unroll
    for (int r = 0; r < ROWS_PER_WAVE; r += 2) {
        float p0 = 0.0f, p1 = 0.0f;

        const int i0 = i_base + r;
        if (i0 < imax) {                 // wave-uniform branch (EXEC stays full)
            const float ideal = floorf((float)i0 / fout * fin);
            const int   j     = (int)ideal - 15 + lane;   // 32-wide exact band
            const float d     = (float)j - ideal;
            const float g     = __expf(-d * d * INV_2SIG2);
            float v = 0.0f;
            if (j >= 0 && j < il) v = row0[(size_t)i0 * E_DIM + j];
            p0 = v * g;
        }
        const int i1 = i0 + 1;
        if (i1 < imax) {
            const float ideal = floorf((float)i1 / fout * fin);
            const int   j     = (int)ideal - 15 + lane;
            const float d     = (float)j - ideal;
            const float g     = __expf(-d * d * INV_2SIG2);
            float v = 0.0f;
            if (j >= 0 && j < il) v = row0[(size_t)i1 * E_DIM + j];
            p1 = v * g;
        }

        v2f a; a.x = p0; a.y = p1;       // A fragment: 2 f32 per lane (16x4 tile)
        // D = A x ones + C : every fed value accumulates (16x redundantly).
        c = __builtin_amdgcn_wmma_f32_16x16x4_f32(
                /*neg_a=*/false, a, /*neg_b=*/false, ones,
                /*c_mod=*/(short)0, c, /*reuse_a=*/false, /*reuse_b=*/false);
    }

    // sum(C over all lanes/regs) == 16 * (wave's true sum)
    float s = c[0] + c[1] + c[2] + c[3] + c[4] + c[5] + c[6] + c[7];
    #pragma unroll
    for (int off = 16; off > 0; off >>= 1)
        s += __shfl_down(s, off, 32);

    __shared__ float smem[WAVES_PER_BLK];
    if (lane == 0) smem[wave] = s;
    __syncthreads();
    if (threadIdx.x == 0) {
        float t = 0.0f;
        #pragma unroll
        for (int w = 0; w < WAVES_PER_BLK; ++w) t += smem[w];
        partials[(size_t)b * GRID_X + blockIdx.x] = t * (1.0f / 16.0f);
    }
}

__global__ __launch_bounds__(256)
void ga_finalize_kernel(const float* __restrict__ partials, float* __restrict__ out)
{
    float s = 0.0f;
    for (int idx = threadIdx.x; idx < NPART; idx += 256)
        s += partials[idx];

    #pragma unroll
    for (int off = 16; off > 0; off >>= 1)
        s += __shfl_down(s, off, 32);

    __shared__ float smem[8];
    const int lane = threadIdx.x & 31;
    const int wave = threadIdx.x >> 5;
    if (lane == 0) smem[wave] = s;
    __syncthreads();
    if (threadIdx.x == 0) {
        float t = 0.0f;
        #pragma unroll
        for (int w = 0; w < 8; ++w) t += smem[w];
        out[0] = t / N_TOTAL;
    }
}

extern "C" void kernel_launch(void* const* d_in, const int* in_sizes, int n_in,
                              void* d_out, int out_size, void* d_ws, size_t ws_size,
                              hipStream_t stream)
{
    const float* aw      = (const float*)d_in[0];
    const int*   in_len  = (const int*)d_in[1];   // input_lengths
    const int*   out_len = (const int*)d_in[2];   // output_lengths
    float*       out     = (float*)d_out;
    float*       part    = (float*)d_ws;          // NPART floats of scratch

    dim3 grid(GRID_X, B_DIM, 1);
    ga_band_sum_kernel<<<grid, 256, 0, stream>>>(aw, in_len, out_len, part);
    ga_finalize_kernel<<<1, 256, 0, stream>>>(part, out);
}